// OuterProductMean_66520453481107
// MI455X (gfx1250) — compile-verified
//
#include <hip/hip_runtime.h>

#define S_ 256
#define N_ 768
#define CM_ 256
#define CH_ 32
#define CZ_ 128
#define LN_EPS 1e-5f

typedef float v2f __attribute__((ext_vector_type(2)));
typedef float v8f __attribute__((ext_vector_type(8)));
typedef unsigned int v4u __attribute__((ext_vector_type(4)));
typedef int v4i __attribute__((ext_vector_type(4)));
typedef int v8i __attribute__((ext_vector_type(8)));

__device__ __forceinline__ float lane_bcast(float v, int src) {
    return __shfl(v, src, 32);
}

// ---------------------------------------------------------------------------
// Kernel 0: zero pooled accumulator, compute 1/max(sum(msa_mask),1)
// ---------------------------------------------------------------------------
__global__ __launch_bounds__(256) void opm_init(
    const unsigned char* __restrict__ msa_mask,
    float* __restrict__ Pacc,         // [N_,64]
    float* __restrict__ denomInv)     // [1]
{
    const int tid = threadIdx.x;
    for (int i = tid; i < N_ * 64; i += 256) Pacc[i] = 0.f;
    __shared__ float red[256];
    red[tid] = (tid < S_) ? (msa_mask[tid] ? 1.f : 0.f) : 0.f;
    __syncthreads();
    for (int w = 128; w > 0; w >>= 1) {
        if (tid < w) red[tid] += red[tid + w];
        __syncthreads();
    }
    if (tid == 0) denomInv[0] = 1.f / fmaxf(red[0], 1.f);
}

// ---------------------------------------------------------------------------
// Kernel 1: fused LayerNorm + left/right projection (WMMA) + masked pool.
// grid = N_*2 blocks of 256 threads (8 waves). Block stages a 128(s) x 256(c)
// f32 tile for one column n with a single Tensor-Data-Mover load (HW-padded
// LDS rows, stride 260 dwords -> conflict-free fragment reads). Each wave
// then LayerNorms its 16-row sub-tile and runs 64x4 chained f32 WMMAs against
// [W_left | W_right], finishing with a masked row-reduction into Pacc[n][64].
// ---------------------------------------------------------------------------
__global__ __launch_bounds__(256) void opm_pool(
    const float* __restrict__ m,              // [S_,N_,CM_]
    const unsigned char* __restrict__ msa_mask,
    const float* __restrict__ gamma,
    const float* __restrict__ beta,
    const float* __restrict__ Wl,             // [CM_,CH_]
    const float* __restrict__ bl,
    const float* __restrict__ Wr,             // [CM_,CH_]
    const float* __restrict__ br,
    float* __restrict__ Pacc)                 // [N_,64]
{
    extern __shared__ float smem[];           // 128*260 + 512 floats
    const int tid  = threadIdx.x;
    const int lane = tid & 31;
    const int wv   = tid >> 5;
    const int n    = blockIdx.x >> 1;
    const int half = blockIdx.x & 1;
    const int s0   = (half * 8 + wv) * 16;    // 16 MSA rows per wave

    float* xs = smem + wv * (16 * 260);       // HW-padded row stride: 260 dwords
    float* gs = smem + 128 * 260;             // gamma [256]
    float* bs = gs + 256;                     // beta  [256]
    gs[tid] = gamma[tid];
    bs[tid] = beta[tid];

    // ---- TDM: one tensor_load_to_lds stages the whole 128x256 f32 tile ----
    if (wv == 0) {
        const unsigned long long gaddr =
            (unsigned long long)(const void*)(m + ((size_t)(half * 128) * N_ + n) * (size_t)CM_);
        v4u g0;
        g0.x = 1u;                                            // count=1 (valid user D#)
        g0.y = 0u;                                            // lds_addr = 0 (dyn-LDS base)
        g0.z = (unsigned int)(gaddr & 0xffffffffu);           // global_addr[31:0]
        g0.w = (unsigned int)((gaddr >> 32) & 0x01ffffffu)    // global_addr[56:32]
             | (2u << 30);                                    // type = 2 ("image")
        v8i g1;
        g1[0] = (int)((2u << 16)      // data_size = 4 bytes
                    | (1u << 20)      // pad_enable
                    | (7u << 22)      // pad_interval: every 256 DWORDs
                    | (3u << 25));    // pad_amount: 4 DWORDs  -> row stride 260
        g1[1] = (int)(256u << 16);    // tensor_dim0 = 256 (bits 79:48)
        g1[2] = (int)(128u << 16);    // tensor_dim1 = 128 (bits 111:80)
        g1[3] = (int)(256u << 16);    // tile_dim0   = 256 (bits 127:112)
        g1[4] = 128;                  // tile_dim1   = 128 (bits 143:128)
        g1[5] = (int)(N_ * CM_);      // tensor_dim0_stride = 196608 (bits 207:160)
        g1[6] = 0;
        g1[7] = 0;
        v4i gz4 = {0, 0, 0, 0};       // groups 2/3 unused (2-D tensor)
        v8i gz8 = {0, 0, 0, 0, 0, 0, 0, 0};
        __builtin_amdgcn_tensor_load_to_lds(g0, g1, gz4, gz4, gz8, 0);
        __builtin_amdgcn_s_wait_tensorcnt(0);
    }
    __syncthreads();

    const int hi  = lane >> 4;    // half-select (K-pair / row-group)
    const int col = lane & 15;
    const int r   = lane & 15;    // this lane's LN row

    // ---- LayerNorm stats: lane covers half the channels of row r ----
    float su = 0.f, sq = 0.f;
    const float4* rp = (const float4*)(xs + r * 260 + hi * 128);
    #pragma unroll
    for (int t = 0; t < 32; ++t) {
        float4 v = rp[t];
        su += v.x + v.y + v.z + v.w;
        sq += v.x * v.x + v.y * v.y + v.z * v.z + v.w * v.w;
    }
    su += __shfl_xor(su, 16, 32);
    sq += __shfl_xor(sq, 16, 32);
    const float mu = su * (1.f / CM_);
    const float rs = rsqrtf(sq * (1.f / CM_) - mu * mu + LN_EPS);

    // ---- row masks for this lane's D-fragment rows ----
    float mk[8];
    float cnt = 0.f;
    #pragma unroll
    for (int v = 0; v < 8; ++v) {
        mk[v] = msa_mask[s0 + v + 8 * hi] ? 1.f : 0.f;
        cnt += mk[v];
    }
    cnt += __shfl_xor(cnt, 16, 32);

    // ---- K-loop: 64 chunks of K=4; 4 output column tiles (64 channels) ----
    v8f acc0 = {}, acc1 = {}, acc2 = {}, acc3 = {};
    #pragma unroll 4
    for (int kk = 0; kk < 64; ++kk) {
        const int c0 = kk * 4 + 2 * hi;
        v2f xv = *(const v2f*)(xs + r * 260 + c0);
        v2f a;
        a.x = (xv.x - mu) * rs * gs[c0]     + bs[c0];
        a.y = (xv.y - mu) * rs * gs[c0 + 1] + bs[c0 + 1];

        v2f b0, b1, b2, b3;
        b0.x = Wl[c0 * CH_ + col];           b0.y = Wl[(c0 + 1) * CH_ + col];
        b1.x = Wl[c0 * CH_ + 16 + col];      b1.y = Wl[(c0 + 1) * CH_ + 16 + col];
        b2.x = Wr[c0 * CH_ + col];           b2.y = Wr[(c0 + 1) * CH_ + col];
        b3.x = Wr[c0 * CH_ + 16 + col];      b3.y = Wr[(c0 + 1) * CH_ + 16 + col];

        acc0 = __builtin_amdgcn_wmma_f32_16x16x4_f32(false, a, false, b0, (short)0, acc0, false, false);
        acc1 = __builtin_amdgcn_wmma_f32_16x16x4_f32(false, a, false, b1, (short)0, acc1, false, false);
        acc2 = __builtin_amdgcn_wmma_f32_16x16x4_f32(false, a, false, b2, (short)0, acc2, false, false);
        acc3 = __builtin_amdgcn_wmma_f32_16x16x4_f32(false, a, false, b3, (short)0, acc3, false, false);
    }

    // ---- masked row-reduction of D fragments; add bias*count; atomic add ----
    #pragma unroll
    for (int ct = 0; ct < 4; ++ct) {
        const v8f* ap = (ct == 0) ? &acc0 : (ct == 1) ? &acc1 : (ct == 2) ? &acc2 : &acc3;
        float p = 0.f;
        #pragma unroll
        for (int v = 0; v < 8; ++v) p += (*ap)[v] * mk[v];
        p += __shfl_xor(p, 16, 32);
        const int hsub = (ct & 1) * 16 + col;                 // 0..31 within matrix
        const float bh = (ct < 2) ? bl[hsub] : br[hsub];
        p += bh * cnt;
        if (hi == 0) atomicAdd(&Pacc[n * 64 + ct * 16 + col], p);
    }
}

// ---------------------------------------------------------------------------
// Kernel 2: z[i,j,k] = mask(i,j) * ( sum_c l[i,c]*r[j,c]*Wout[c,k] + bout[k] )
// Wave job = (k_tile, i_tile, j_group of 16). 8 chained f32 WMMAs per j.
// ---------------------------------------------------------------------------
__global__ __launch_bounds__(256) void opm_outer(
    const float* __restrict__ Pacc,           // [N_,64]: [:,0:32]=l sums, [:,32:64]=r sums
    const float* __restrict__ denomInv,
    const unsigned char* __restrict__ residue_mask,
    const float* __restrict__ Wout,           // [CH_,CZ_]
    const float* __restrict__ bout,           // [CZ_]
    float* __restrict__ z)                    // [N_,N_,CZ_]
{
    const int lane = threadIdx.x & 31;
    int job = blockIdx.x * 8 + (threadIdx.x >> 5);
    const int NKT = CZ_ / 16;                 // 8
    const int NIT = N_ / 16;                  // 48
    const int kt = job % NKT;  job /= NKT;
    const int it = job % NIT;  job /= NIT;
    const int jg = job;                       // 0..47
    const int k0 = kt * 16, i0 = it * 16, j0 = jg * 16;

    const int hi  = lane >> 4;
    const int col = lane & 15;
    const float inv = denomInv[0];

    // A fragments: l tile (16 x 32), scaled by 1/denom; held across 16 j's
    v2f a[8];
    const float* lrow = Pacc + (size_t)(i0 + col) * 64;
    #pragma unroll
    for (int kk = 0; kk < 8; ++kk) {
        const int cb = kk * 4 + 2 * hi;
        a[kk].x = lrow[cb]     * inv;
        a[kk].y = lrow[cb + 1] * inv;
    }
    // Wout tile (32 x 16) in registers
    float w0[8], w1[8];
    #pragma unroll
    for (int kk = 0; kk < 8; ++kk) {
        const int cb = kk * 4 + 2 * hi;
        w0[kk] = Wout[cb * CZ_ + k0 + col];
        w1[kk] = Wout[(cb + 1) * CZ_ + k0 + col];
    }
    const float bk = bout[k0 + col];
    float mrow[8];
    #pragma unroll
    for (int v = 0; v < 8; ++v)
        mrow[v] = residue_mask[i0 + v + 8 * hi] ? 1.f : 0.f;

    for (int j = j0; j < j0 + 16; ++j) {
        const float rv = Pacc[j * 64 + 32 + lane] * inv;  // lane holds r[j, c=lane]
        const float mj = residue_mask[j] ? 1.f : 0.f;
        v8f c8 = {};
        #pragma unroll
        for (int kk = 0; kk < 8; ++kk) {
            const int cb = kk * 4 + 2 * hi;
            v2f b;
            b.x = w0[kk] * lane_bcast(rv, cb);
            b.y = w1[kk] * lane_bcast(rv, cb + 1);
            c8 = __builtin_amdgcn_wmma_f32_16x16x4_f32(false, a[kk], false, b, (short)0, c8, false, false);
        }
        float* zp = z + ((size_t)(i0 + 8 * hi) * N_ + j) * CZ_ + k0 + col;
        #pragma unroll
        for (int v = 0; v < 8; ++v)
            zp[(size_t)v * N_ * CZ_] = (c8[v] + bk) * mrow[v] * mj;
    }
}

// ---------------------------------------------------------------------------
extern "C" void kernel_launch(void* const* d_in, const int* in_sizes, int n_in,
                              void* d_out, int out_size, void* d_ws, size_t ws_size,
                              hipStream_t stream) {
    const float*         m     = (const float*)d_in[0];
    const unsigned char* msa   = (const unsigned char*)d_in[1];
    const unsigned char* rmask = (const unsigned char*)d_in[2];
    const float*         gam   = (const float*)d_in[3];
    const float*         bet   = (const float*)d_in[4];
    const float*         Wl    = (const float*)d_in[5];
    const float*         bl    = (const float*)d_in[6];
    const float*         Wr    = (const float*)d_in[7];
    const float*         br    = (const float*)d_in[8];
    const float*         Wout  = (const float*)d_in[9];
    const float*         bout  = (const float*)d_in[10];
    float*               zout  = (float*)d_out;

    float* Pacc     = (float*)d_ws;           // [N_*64]
    float* denomInv = Pacc + N_ * 64;         // [1]

    opm_init<<<1, 256, 0, stream>>>(msa, Pacc, denomInv);

    const size_t lds_pool = (128 * 260 + 512) * sizeof(float);  // 135168 B
    opm_pool<<<N_ * 2, 256, lds_pool, stream>>>(m, msa, gam, bet, Wl, bl, Wr, br, Pacc);

    const int jobs   = (CZ_ / 16) * (N_ / 16) * (N_ / 16);  // 8*48*48 = 18432 waves
    const int blocks = jobs / 8;                            // 2304
    opm_outer<<<blocks, 256, 0, stream>>>(Pacc, denomInv, rmask, Wout, bout, zout);
}